// DiscTransformerLayer_14267881357664
// MI455X (gfx1250) — compile-verified
//
#include <hip/hip_runtime.h>
#include <math.h>

#define TT 8
#define THH 16
#define TWW 24
#define NTOK 3072        // 8*16*24
#define BB 2
#define DMODEL 512
#define NHEAD 8
#define HDIM 64
#define EDIM 2048
#define MROWS (BB * NTOK)   // 6144

typedef __attribute__((ext_vector_type(2))) float f2;
typedef __attribute__((ext_vector_type(8))) float f8;

// epilogue mode bits
#define EP_BIAS  1
#define EP_GELU  2
#define EP_RESID 4
#define EP_MASK  8

// ---------------------------------------------------------------------------
// LayerNorm: one 256-thread block per row of 512
// ---------------------------------------------------------------------------
__global__ __launch_bounds__(256)
void ln_kernel(const float* __restrict__ x, const float* __restrict__ g,
               const float* __restrict__ be, float* __restrict__ out)
{
    const int row = blockIdx.x;
    const float* xr = x + (size_t)row * DMODEL;
    const int tid = threadIdx.x;

    float v0 = xr[tid];
    float v1 = xr[tid + 256];
    float s  = v0 + v1;
    float ss = v0 * v0 + v1 * v1;

#pragma unroll
    for (int off = 16; off; off >>= 1) {
        s  += __shfl_xor(s, off, 32);
        ss += __shfl_xor(ss, off, 32);
    }

    __shared__ float sm[8], sm2[8];
    const int wid = tid >> 5, lane = tid & 31;
    if (lane == 0) { sm[wid] = s; sm2[wid] = ss; }
    __syncthreads();
    if (tid == 0) {
        float ts = 0.f, tss = 0.f;
#pragma unroll
        for (int i = 0; i < 8; ++i) { ts += sm[i]; tss += sm2[i]; }
        sm[0] = ts; sm2[0] = tss;
    }
    __syncthreads();

    const float mean = sm[0] * (1.f / DMODEL);
    const float var  = sm2[0] * (1.f / DMODEL) - mean * mean;
    const float rstd = rsqrtf(var + 1e-5f);

    float* orow = out + (size_t)row * DMODEL;
    orow[tid]       = (v0 - mean) * rstd * g[tid]       + be[tid];
    orow[tid + 256] = (v1 - mean) * rstd * g[tid + 256] + be[tid + 256];
}

// ---------------------------------------------------------------------------
// Generic fp32 WMMA GEMM: C[M x Nn] = A[M x K] @ B[K x Nn]  (+epilogue)
// grid = (Nn/64, M/32), block = 32.
// One wave owns a 32x64 tile: 2 A-fragments x 4 B-fragments -> 8 WMMA per
// K-step of 4, so each B fragment (weights, L2-resident) is reused twice.
// ---------------------------------------------------------------------------
__global__ __launch_bounds__(32)
void gemm_wmma_f32(const float* __restrict__ A, const float* __restrict__ Bm,
                   float* __restrict__ C, const float* __restrict__ bias,
                   const float* __restrict__ resid,
                   const unsigned char* __restrict__ mask,
                   int M, int K, int Nn, int mode)
{
    const int n0   = blockIdx.x * 64;
    const int m0   = blockIdx.y * 32;
    const int lane = threadIdx.x & 31;
    const int half = lane >> 4;     // 0: K pair {0,1}, 1: K pair {2,3}
    const int l16  = lane & 15;     // A row / B col within tile

    f8 acc[2][4];
#pragma unroll
    for (int i = 0; i < 2; ++i)
#pragma unroll
        for (int j = 0; j < 4; ++j) acc[i][j] = 0.f;

    const float* Arow0 = A + (size_t)(m0 + l16) * K;
    const float* Arow1 = Arow0 + (size_t)16 * K;

    for (int k0 = 0; k0 < K; k0 += 4) {
        const int kk = k0 + 2 * half;
        f2 a0, a1;
        a0.x = Arow0[kk];
        a0.y = Arow0[kk + 1];
        a1.x = Arow1[kk];
        a1.y = Arow1[kk + 1];
        const float* bp = Bm + (size_t)kk * Nn + n0 + l16;
#pragma unroll
        for (int j = 0; j < 4; ++j) {
            f2 b;
            b.x = bp[16 * j];
            b.y = bp[16 * j + Nn];
            acc[0][j] = __builtin_amdgcn_wmma_f32_16x16x4_f32(
                false, a0, false, b, (short)0, acc[0][j], false, false);
            acc[1][j] = __builtin_amdgcn_wmma_f32_16x16x4_f32(
                false, a1, false, b, (short)0, acc[1][j], false, false);
        }
    }

#pragma unroll
    for (int i = 0; i < 2; ++i) {
#pragma unroll
        for (int j = 0; j < 4; ++j) {
            const int col = n0 + 16 * j + l16;
            const float bv = (mode & EP_BIAS) ? bias[col] : 0.f;
#pragma unroll
            for (int r = 0; r < 8; ++r) {
                const int row = m0 + 16 * i + r + 8 * half;
                float v = acc[i][j][r] + bv;
                if (mode & EP_GELU)
                    v = 0.5f * v * (1.f + erff(v * 0.70710678118654752f));
                if (mode & EP_MASK)
                    v = mask[row] ? v : 0.f;
                if (mode & EP_RESID)
                    v += resid[(size_t)row * Nn + col];
                C[(size_t)row * Nn + col] = v;
            }
        }
    }
}

// ---------------------------------------------------------------------------
// Local 3x3x3 neighborhood attention. grid = B*N, block = 256 (wave = head).
// qkv layout per row: [0,512)=q [512,1024)=k [1024,1536)=v
// ---------------------------------------------------------------------------
__global__ __launch_bounds__(256)
void local_attn_kernel(const float* __restrict__ qkv, float* __restrict__ attn)
{
    const int bn   = blockIdx.x;
    const int b    = bn / NTOK;
    const int n    = bn % NTOK;
    const int head = threadIdx.x >> 5;
    const int lane = threadIdx.x & 31;

    const int t   = n / (THH * TWW);
    const int rem = n % (THH * TWW);
    const int hh  = rem / TWW;
    const int ww  = rem % TWW;

    const float* base = qkv + (size_t)bn * 1536 + head * HDIM + 2 * lane;
    f2 q;
    q.x = base[0] * 0.125f;   // 1/sqrt(64)
    q.y = base[1] * 0.125f;

    float s[27];
#pragma unroll
    for (int j = 0; j < 27; ++j) {
        const int dt = j / 9 - 1, dh = (j / 3) % 3 - 1, dw = j % 3 - 1;
        const int tt = t + dt, h2 = hh + dh, w2 = ww + dw;
        const bool valid = (tt >= 0 && tt < TT && h2 >= 0 && h2 < THH &&
                            w2 >= 0 && w2 < TWW);
        float px = 0.f;
        if (valid) {
            const int nn = (tt * THH + h2) * TWW + w2;
            const float* kp = qkv + (size_t)(b * NTOK + nn) * 1536 + 512 +
                              head * HDIM + 2 * lane;
            px = q.x * kp[0] + q.y * kp[1];
        }
#pragma unroll
        for (int off = 16; off; off >>= 1) px += __shfl_xor(px, off, 32);
        s[j] = px;   // zero-padded neighbors contribute q.0 = 0 exactly
    }

    float m = s[0];
#pragma unroll
    for (int j = 1; j < 27; ++j) m = fmaxf(m, s[j]);
    float denom = 0.f;
    float p[27];
#pragma unroll
    for (int j = 0; j < 27; ++j) { p[j] = __expf(s[j] - m); denom += p[j]; }

    f2 o; o.x = 0.f; o.y = 0.f;
#pragma unroll
    for (int j = 0; j < 27; ++j) {
        const int dt = j / 9 - 1, dh = (j / 3) % 3 - 1, dw = j % 3 - 1;
        const int tt = t + dt, h2 = hh + dh, w2 = ww + dw;
        if (tt >= 0 && tt < TT && h2 >= 0 && h2 < THH && w2 >= 0 && w2 < TWW) {
            const int nn = (tt * THH + h2) * TWW + w2;
            const float* vp = qkv + (size_t)(b * NTOK + nn) * 1536 + 1024 +
                              head * HDIM + 2 * lane;
            o.x += p[j] * vp[0];
            o.y += p[j] * vp[1];
        }
    }
    const float rl = 1.f / denom;
    float* op = attn + (size_t)bn * DMODEL + head * HDIM + 2 * lane;
    op[0] = o.x * rl;
    op[1] = o.y * rl;
}

// ---------------------------------------------------------------------------
// Global attention, flash-style. grid = B*H*(N/16), block = 32 (one wave).
// Computes S^T = K_tile x Q^T so each query's softmax lives in lanes (q,q+16),
// then O^T += V^T x P^T with P^T reshuffled C-layout -> B-fragments.
// ---------------------------------------------------------------------------
__global__ __launch_bounds__(32)
void global_attn_kernel(const float* __restrict__ qkv, float* __restrict__ attn)
{
    const int nqt  = NTOK / 16;                 // 192
    const int blk  = blockIdx.x;
    const int qt   = blk % nqt;
    const int tmp  = blk / nqt;
    const int head = tmp % NHEAD;
    const int b    = tmp / NHEAD;

    const int lane = threadIdx.x & 31;
    const int half = lane >> 4;
    const int l16  = lane & 15;

    // Q tile as B-fragments of Q^T (pre-scaled by 1/sqrt(d))
    const float* qbase =
        qkv + (size_t)(b * NTOK + qt * 16 + l16) * 1536 + head * HDIM;
    f2 qb[16];
#pragma unroll
    for (int i = 0; i < 16; ++i) {
        const int kk = 4 * i + 2 * half;
        qb[i].x = qbase[kk] * 0.125f;
        qb[i].y = qbase[kk + 1] * 0.125f;
    }

    f8 acc[4];
#pragma unroll
    for (int c = 0; c < 4; ++c) acc[c] = 0.f;
    float mrun = -3.0e38f;
    float lrun = 0.f;

    for (int kt = 0; kt < nqt; ++kt) {
        // S^T tile (16 keys x 16 queries)
        const float* kbase =
            qkv + (size_t)(b * NTOK + kt * 16 + l16) * 1536 + 512 + head * HDIM;
        f8 st = 0.f;
#pragma unroll
        for (int i = 0; i < 16; ++i) {
            const int kk = 4 * i + 2 * half;
            f2 a;
            a.x = kbase[kk];
            a.y = kbase[kk + 1];
            st = __builtin_amdgcn_wmma_f32_16x16x4_f32(
                false, a, false, qb[i], (short)0, st, false, false);
        }

        // per-query (column) max over this tile: 8 in-lane + xor16
        float tm = st[0];
#pragma unroll
        for (int r = 1; r < 8; ++r) tm = fmaxf(tm, st[r]);
        tm = fmaxf(tm, __shfl_xor(tm, 16, 32));

        const float mnew  = fmaxf(mrun, tm);
        const float scale = __expf(mrun - mnew);

        f8 p;
        float ts = 0.f;
#pragma unroll
        for (int r = 0; r < 8; ++r) { p[r] = __expf(st[r] - mnew); ts += p[r]; }
        ts += __shfl_xor(ts, 16, 32);
        lrun = lrun * scale + ts;
        mrun = mnew;

#pragma unroll
        for (int c = 0; c < 4; ++c)
#pragma unroll
            for (int r = 0; r < 8; ++r) acc[c][r] *= scale;

        // reshuffle P^T C-layout into B-fragments (rows 4j+2h, 4j+2h+1)
        float sp[8];
#pragma unroll
        for (int r = 0; r < 8; ++r) sp[r] = __shfl_xor(p[r], 16, 32);
        f2 pb[4];
        if (half == 0) {
            pb[0].x = p[0];  pb[0].y = p[1];
            pb[1].x = p[4];  pb[1].y = p[5];
            pb[2].x = sp[0]; pb[2].y = sp[1];
            pb[3].x = sp[4]; pb[3].y = sp[5];
        } else {
            pb[0].x = sp[2]; pb[0].y = sp[3];
            pb[1].x = sp[6]; pb[1].y = sp[7];
            pb[2].x = p[2];  pb[2].y = p[3];
            pb[3].x = p[6];  pb[3].y = p[7];
        }

        // O^T (4 chunks of 16 dims x 16 queries) += V^T x P^T
#pragma unroll
        for (int c = 0; c < 4; ++c) {
#pragma unroll
            for (int j = 0; j < 4; ++j) {
                const int key = kt * 16 + j * 4 + 2 * half;
                const float* vp = qkv + (size_t)(b * NTOK + key) * 1536 + 1024 +
                                  head * HDIM + 16 * c + l16;
                f2 a;
                a.x = vp[0];
                a.y = vp[1536];
                acc[c] = __builtin_amdgcn_wmma_f32_16x16x4_f32(
                    false, a, false, pb[j], (short)0, acc[c], false, false);
            }
        }
    }

    const float rl = 1.f / lrun;
#pragma unroll
    for (int c = 0; c < 4; ++c) {
#pragma unroll
        for (int r = 0; r < 8; ++r) {
            const int d = 16 * c + r + 8 * half;
            attn[(size_t)(b * NTOK + qt * 16 + l16) * DMODEL + head * HDIM + d] =
                acc[c][r] * rl;
        }
    }
}

// ---------------------------------------------------------------------------
extern "C" void kernel_launch(void* const* d_in, const int* in_sizes, int n_in,
                              void* d_out, int out_size, void* d_ws,
                              size_t ws_size, hipStream_t stream)
{
    const float* x      = (const float*)d_in[0];
    const unsigned char* mask = (const unsigned char*)d_in[1];
    const float* ln_l_g = (const float*)d_in[2];
    const float* ln_l_b = (const float*)d_in[3];
    const float* ln_g_g = (const float*)d_in[4];
    const float* ln_g_b = (const float*)d_in[5];
    const float* ln_f_g = (const float*)d_in[6];
    const float* ln_f_b = (const float*)d_in[7];
    const float* Wqkv_l = (const float*)d_in[8];
    const float* Wo_l   = (const float*)d_in[9];
    const float* Wqkv_g = (const float*)d_in[10];
    const float* Wo_g   = (const float*)d_in[11];
    const float* W1     = (const float*)d_in[12];
    const float* b1     = (const float*)d_in[13];
    const float* W2     = (const float*)d_in[14];
    const float* b2     = (const float*)d_in[15];

    float* out = (float*)d_out;
    float* ws  = (float*)d_ws;

    float* xn   = ws;                                 // MROWS * 512
    float* qkv  = xn + (size_t)MROWS * DMODEL;        // MROWS * 2048 (qkv / ffn h)
    float* attn = qkv + (size_t)MROWS * EDIM;         // MROWS * 512

    const dim3 g_qkv(3 * DMODEL / 64, MROWS / 32);
    const dim3 g_d(DMODEL / 64, MROWS / 32);
    const dim3 g_e(EDIM / 64, MROWS / 32);

    // ---- sublayer 1: local attention ----
    ln_kernel<<<MROWS, 256, 0, stream>>>(x, ln_l_g, ln_l_b, xn);
    gemm_wmma_f32<<<g_qkv, 32, 0, stream>>>(xn, Wqkv_l, qkv, nullptr, nullptr,
                                            nullptr, MROWS, DMODEL, 3 * DMODEL, 0);
    local_attn_kernel<<<MROWS, 256, 0, stream>>>(qkv, attn);
    gemm_wmma_f32<<<g_d, 32, 0, stream>>>(attn, Wo_l, out, nullptr, x, nullptr,
                                          MROWS, DMODEL, DMODEL, EP_RESID);

    // ---- sublayer 2: global (masked) attention ----
    ln_kernel<<<MROWS, 256, 0, stream>>>(out, ln_g_g, ln_g_b, xn);
    gemm_wmma_f32<<<g_qkv, 32, 0, stream>>>(xn, Wqkv_g, qkv, nullptr, nullptr,
                                            nullptr, MROWS, DMODEL, 3 * DMODEL, 0);
    global_attn_kernel<<<BB * NHEAD * (NTOK / 16), 32, 0, stream>>>(qkv, attn);
    gemm_wmma_f32<<<g_d, 32, 0, stream>>>(attn, Wo_g, out, nullptr, out, mask,
                                          MROWS, DMODEL, DMODEL,
                                          EP_RESID | EP_MASK);

    // ---- sublayer 3: FFN ----
    ln_kernel<<<MROWS, 256, 0, stream>>>(out, ln_f_g, ln_f_b, xn);
    gemm_wmma_f32<<<g_e, 32, 0, stream>>>(xn, W1, qkv, b1, nullptr, nullptr,
                                          MROWS, DMODEL, EDIM, EP_BIAS | EP_GELU);
    gemm_wmma_f32<<<g_d, 32, 0, stream>>>(qkv, W2, out, b2, out, nullptr,
                                          MROWS, EDIM, DMODEL,
                                          EP_BIAS | EP_RESID);
}